// graphNetwork_46746424050473
// MI455X (gfx1250) — compile-verified
//
#include <hip/hip_runtime.h>

// ---------------------------------------------------------------------------
// CDNA5 (gfx1250) implementation of the graph-network reference.
// fp32 end-to-end using V_WMMA_F32_16X16X4_F32: the workload is HBM-bound
// (~2GB of tensor traffic @ 23.3 TB/s vs ~35 GFLOP of GEMM), so fp32 matrix
// ops keep exact precision at zero speed cost.  All layernorm+relu passes are
// fused into the consumer GEMM's B-load path; verlet updates write final xe
// directly to d_out; anchors are produced in place (no d2d copies).
// ---------------------------------------------------------------------------

typedef __attribute__((ext_vector_type(2))) float v2f;
typedef __attribute__((ext_vector_type(8))) float v8f;

__device__ __forceinline__ v8f wmma_f32_16x16x4(v2f a, v2f b, v8f c) {
  // D = A(16x4) * B(4x16) + C(16x16), all fp32, wave32.
  return __builtin_amdgcn_wmma_f32_16x16x4_f32(
      /*neg_a=*/false, a, /*neg_b=*/false, b,
      /*c_mod=*/(short)0, c, /*reuse_a=*/false, /*reuse_b=*/false);
}

__device__ __forceinline__ void wave_sum_atomic(float s, float sq, float* red) {
#pragma unroll
  for (int off = 16; off > 0; off >>= 1) {
    s  += __shfl_down(s, off, 32);
    sq += __shfl_down(sq, off, 32);
  }
  if ((threadIdx.x & 31) == 0) {
    atomicAdd(&red[0], s);
    atomicAdd(&red[1], sq);
  }
}

// ---------------------------------------------------------------------------
// Generic GEMM: Y[O,N] = W[O,C] @ LNReLU?(X[C,N]).  O = OT*16, C%4==0, N%16==0.
// Weights staged in LDS with +1 padding (conflict-free A reads).
// redIn  (optional): (sum,sum^2) of X -> apply global layernorm + relu to B.
// redOut (optional): accumulate (sum,sum^2) of Y for a following layernorm.
// ---------------------------------------------------------------------------
template <int OT>
__global__ void __launch_bounds__(256)
gemm_ln_kernel(const float* __restrict__ W, const float* __restrict__ X,
               float* __restrict__ Y, int C, long N, float* redOut,
               const float* __restrict__ redIn, float invCntIn) {
  extern __shared__ float Ws[];  // [OT*16][C+1]
  const int tid = threadIdx.x;
  const int O = OT * 16;
  const int Cp = C + 1;
  for (int t = tid; t < O * C; t += 256) Ws[(t / C) * Cp + (t % C)] = W[t];
  __syncthreads();

  const int w = tid >> 5, l = tid & 31;
  const int half = l >> 4, lane16 = l & 15;
  const long n0 = (long)blockIdx.x * 128 + (long)w * 16;
  if (n0 >= N) return;

  float mean = 0.f, rstd = 1.f;
  const bool norm = (redIn != nullptr);
  if (norm) {
    mean = redIn[0] * invCntIn;
    const float var = redIn[1] * invCntIn - mean * mean;
    rstd = rsqrtf(var + 1e-5f);
  }

  v8f zero;
#pragma unroll
  for (int r = 0; r < 8; ++r) zero[r] = 0.f;
  v8f acc[OT];
#pragma unroll
  for (int t = 0; t < OT; ++t) acc[t] = zero;

  const float* Xb = X + n0 + lane16;
  const int k2h = 2 * half;
  for (int k = 0; k < C; k += 4) {
    if (k + 16 < C) __builtin_prefetch(&Xb[(long)(k + 16 + k2h) * N], 0, 1);
    float x0 = Xb[(long)(k + k2h) * N];
    float x1 = Xb[(long)(k + k2h + 1) * N];
    if (norm) {
      x0 = fmaxf(0.f, (x0 - mean) * rstd);
      x1 = fmaxf(0.f, (x1 - mean) * rstd);
    }
    v2f b;
    b.x = x0;
    b.y = x1;
#pragma unroll
    for (int t = 0; t < OT; ++t) {
      const float* Wr = &Ws[(t * 16 + lane16) * Cp + k + k2h];
      v2f a;
      a.x = Wr[0];
      a.y = Wr[1];
      acc[t] = wmma_f32_16x16x4(a, b, acc[t]);
    }
  }

  float s = 0.f, sq = 0.f;
#pragma unroll
  for (int t = 0; t < OT; ++t) {
#pragma unroll
    for (int r = 0; r < 8; ++r) {
      float v = acc[t][r];
      Y[(long)(t * 16 + r + 8 * half) * N + n0 + lane16] = v;
      s += v;
      sq += v * v;
    }
  }
  if (redOut) wave_sum_atomic(s, sq, redOut);
}

// ---------------------------------------------------------------------------
// Fused edge GEMM: hid[64,m] = KE1 @ [intX; xe; gradX] without materializing
// the 307MB concat.  intX/gradX fold into per-endpoint weights:
//   hid = (0.5*A1+A3) @ xn[:,i] + (0.5*A1-A3) @ xn[:,j] + A2 @ xe
// ---------------------------------------------------------------------------
__global__ void __launch_bounds__(256)
gemm_edge_kernel(const float* __restrict__ KE1, const float* __restrict__ xn,
                 const float* __restrict__ xe, const int* __restrict__ iInd,
                 const int* __restrict__ jInd, float* __restrict__ Y, long n,
                 long m, float* redOut) {
  __shared__ float Wi[64 * 65], Wj[64 * 65], We[64 * 65];
  const int tid = threadIdx.x;
  for (int t = tid; t < 64 * 64; t += 256) {
    int r = t >> 6, c = t & 63;
    float a1 = KE1[r * 192 + c];
    float a2 = KE1[r * 192 + 64 + c];
    float a3 = KE1[r * 192 + 128 + c];
    Wi[r * 65 + c] = 0.5f * a1 + a3;
    Wj[r * 65 + c] = 0.5f * a1 - a3;
    We[r * 65 + c] = a2;
  }
  __syncthreads();

  const int w = tid >> 5, l = tid & 31;
  const int half = l >> 4, lane16 = l & 15;
  const long e0 = (long)blockIdx.x * 128 + (long)w * 16;
  if (e0 >= m) return;
  const long col = e0 + lane16;
  const int ii = iInd[col], jj = jInd[col];

  v8f zero;
#pragma unroll
  for (int r = 0; r < 8; ++r) zero[r] = 0.f;
  v8f acc[4];
#pragma unroll
  for (int t = 0; t < 4; ++t) acc[t] = zero;

  const int k2h = 2 * half;
  for (int k = 0; k < 64; k += 4) {
    const int k0 = k + k2h;
    v2f bi, bj, be;
    bi.x = xn[(long)k0 * n + ii];
    bi.y = xn[(long)(k0 + 1) * n + ii];
    bj.x = xn[(long)k0 * n + jj];
    bj.y = xn[(long)(k0 + 1) * n + jj];
    be.x = xe[(long)k0 * m + col];
    be.y = xe[(long)(k0 + 1) * m + col];
#pragma unroll
    for (int t = 0; t < 4; ++t) {
      const int rb = (t * 16 + lane16) * 65 + k0;
      v2f ai, aj, ae;
      ai.x = Wi[rb];
      ai.y = Wi[rb + 1];
      aj.x = Wj[rb];
      aj.y = Wj[rb + 1];
      ae.x = We[rb];
      ae.y = We[rb + 1];
      acc[t] = wmma_f32_16x16x4(ai, bi, acc[t]);
      acc[t] = wmma_f32_16x16x4(aj, bj, acc[t]);
      acc[t] = wmma_f32_16x16x4(ae, be, acc[t]);
    }
  }

  float s = 0.f, sq = 0.f;
#pragma unroll
  for (int t = 0; t < 4; ++t) {
#pragma unroll
    for (int r = 0; r < 8; ++r) {
      float v = acc[t][r];
      Y[(long)(t * 16 + r + 8 * half) * m + col] = v;
      s += v;
      sq += v * v;
    }
  }
  if (redOut) wave_sum_atomic(s, sq, redOut);
}

// ---------------------------------------------------------------------------
// Scatter edges -> node features: rows 0..63 = aveE, rows 64..127 = divE.
// ---------------------------------------------------------------------------
__global__ void __launch_bounds__(256)
scatter_kernel(const float* __restrict__ xe, const int* __restrict__ iInd,
               const int* __restrict__ jInd, float* __restrict__ featN, long n,
               long m) {
  const long e = (long)blockIdx.x * 256 + threadIdx.x;
  if (e >= m) return;
  const int c = blockIdx.y;
  const int ii = iInd[e], jj = jInd[e];
  const float g = xe[(long)c * m + e];
  atomicAdd(&featN[(long)c * n + ii], 0.5f * g);
  atomicAdd(&featN[(long)c * n + jj], 0.5f * g);
  atomicAdd(&featN[(long)(64 + c) * n + ii], g);
  atomicAdd(&featN[(long)(64 + c) * n + jj], -g);
}

// dst = 2a - o + h^2 * (red ? LN(d) : d)
__global__ void __launch_bounds__(256)
verlet_kernel(float* __restrict__ dst, const float* __restrict__ a,
              const float* __restrict__ o, const float* __restrict__ d,
              long size, const float* __restrict__ red, float invCnt) {
  const long t = (long)blockIdx.x * 256 + threadIdx.x;
  if (t >= size) return;
  float v = d[t];
  if (red) {
    const float mean = red[0] * invCnt;
    const float var = red[1] * invCnt - mean * mean;
    v = (v - mean) * rsqrtf(var + 1e-5f);
  }
  dst[t] = 2.f * a[t] - o[t] + 0.01f * v;  // H2 = 0.01
}

// ---------------------------------------------------------------------------
extern "C" void kernel_launch(void* const* d_in, const int* in_sizes, int n_in,
                              void* d_out, int out_size, void* d_ws,
                              size_t ws_size, hipStream_t stream) {
  (void)n_in; (void)out_size; (void)ws_size;
  const float* xn_in = (const float*)d_in[0];
  const float* xe_in = (const float*)d_in[1];
  const int* iInd = (const int*)d_in[2];
  const int* jInd = (const int*)d_in[3];
  const float* K1N = (const float*)d_in[4];
  const float* K2N = (const float*)d_in[5];
  const float* K1E = (const float*)d_in[6];
  const float* K2E = (const float*)d_in[7];
  const float* KNc = (const float*)d_in[8];
  const float* KE1 = (const float*)d_in[9];
  const float* KE2 = (const float*)d_in[10];
  const float* KN1 = (const float*)d_in[11];
  const float* KN2 = (const float*)d_in[12];

  const long nn = in_sizes[0] / 16;             // nodes (40000)
  const long mm = in_sizes[1] / 16;             // edges (400000)
  const int nlayer = in_sizes[9] / (64 * 192);  // 2
  float* out = (float*)d_out;

  float* ws = (float*)d_ws;
  float* red = ws;  // 3 (sum,sumsq) pairs
  float* xn_a = ws + 16;
  float* xn_o = xn_a + 64 * nn;   // opened xn (fixed anchor)
  float* hidN = xn_o + 64 * nn;
  float* dxn = hidN + 64 * nn;
  float* featN = dxn + 64 * nn;   // [192, nn]
  float* xe_a = featN + 192 * nn;
  float* xe_o = xe_a + 64 * mm;   // opened xe (fixed anchor)
  float* hidE = xe_o + 64 * mm;
  float* dxe = hidE + 64 * mm;

  float* redA = red;
  float* redB = red + 2;
  float* redC = red + 4;

  const float icN = 1.f / (float)(64 * nn);
  const float icM = 1.f / (float)(64 * mm);

  auto gblocks = [](long N) { return (unsigned)((N / 16 + 7) / 8); };
  auto eblocks = [](long sz) { return (unsigned)((sz + 255) / 256); };
#define SHB(OT, C) ((size_t)(OT) * 16 * ((C) + 1) * sizeof(float))

  // ---- open node: xn_o = K2N @ LNReLU(K1N @ xn_in) ----
  hipMemsetAsync(redA, 0, 2 * sizeof(float), stream);
  gemm_ln_kernel<4><<<gblocks(nn), 256, SHB(4, 16), stream>>>(
      K1N, xn_in, hidN, 16, nn, redA, nullptr, 0.f);
  gemm_ln_kernel<4><<<gblocks(nn), 256, SHB(4, 64), stream>>>(
      K2N, hidN, xn_o, 64, nn, nullptr, redA, icN);

  // ---- open edge: xe_o = K2E @ LNReLU(K1E @ xe_in) ----
  hipMemsetAsync(redA, 0, 2 * sizeof(float), stream);
  gemm_ln_kernel<4><<<gblocks(mm), 256, SHB(4, 16), stream>>>(
      K1E, xe_in, hidE, 16, mm, redA, nullptr, 0.f);
  gemm_ln_kernel<4><<<gblocks(mm), 256, SHB(4, 64), stream>>>(
      K2E, hidE, xe_o, 64, mm, nullptr, redA, icM);

  const float* xnCur = xn_o;  // layer-0 current state == anchor
  const float* xeCur = xe_o;

  for (int i = 0; i < nlayer; ++i) {
    // node features from pre-update state: [aveE; divE; xn]
    hipMemsetAsync(featN, 0, 128 * nn * sizeof(float), stream);
    scatter_kernel<<<dim3(eblocks(mm), 64), 256, 0, stream>>>(
        xeCur, iInd, jInd, featN, nn, mm);
    hipMemcpyAsync(featN + 128 * nn, xnCur, 64 * nn * sizeof(float),
                   hipMemcpyDeviceToDevice, stream);

    // edge path: hidE = KE1 @ [intX;xe;gradX]; dxe = KE2 @ LNReLU(hidE)
    hipMemsetAsync(redA, 0, 2 * sizeof(float), stream);
    gemm_edge_kernel<<<gblocks(mm), 256, 0, stream>>>(
        KE1 + (long)i * 64 * 192, xnCur, xeCur, iInd, jInd, hidE, nn, mm, redA);
    hipMemsetAsync(redB, 0, 2 * sizeof(float), stream);
    gemm_ln_kernel<4><<<gblocks(mm), 256, SHB(4, 64), stream>>>(
        KE2 + (long)i * 64 * 64, hidE, dxe, 64, mm, redB, redA, icM);

    // node path: hidN = KN1 @ featN; dxn = KN2 @ LNReLU(hidN)
    hipMemsetAsync(redC, 0, 2 * sizeof(float), stream);
    gemm_ln_kernel<4><<<gblocks(nn), 256, SHB(4, 192), stream>>>(
        KN1 + (long)i * 64 * 192, featN, hidN, 192, nn, redC, nullptr, 0.f);
    gemm_ln_kernel<4><<<gblocks(nn), 256, SHB(4, 64), stream>>>(
        KN2 + (long)i * 64 * 64, hidN, dxn, 64, nn, nullptr, redC, icN);

    // verlet updates; last layer streams xe straight into d_out
    float* xeDst = (i == nlayer - 1) ? (out + 32 * nn) : xe_a;
    verlet_kernel<<<eblocks(64 * nn), 256, 0, stream>>>(
        xn_a, xnCur, xn_o, dxn, 64 * nn, nullptr, 0.f);
    verlet_kernel<<<eblocks(64 * mm), 256, 0, stream>>>(
        xeDst, xeCur, xe_o, dxe, 64 * mm, redB, icM);
    xnCur = xn_a;
    xeCur = xeDst;
  }

  if (nlayer == 0) {  // degenerate safety: still populate full output
    hipMemcpyAsync(out + 32 * nn, xe_o, 64 * mm * sizeof(float),
                   hipMemcpyDeviceToDevice, stream);
  }

  // close: out[0:32*nn] = KNclose @ xn
  gemm_ln_kernel<2><<<gblocks(nn), 256, SHB(2, 64), stream>>>(
      KNc, xnCur, out, 64, nn, nullptr, nullptr, 0.f);
#undef SHB
}